// SimpleResBlocksub1F1_86629490361018
// MI455X (gfx1250) — compile-verified
//
#include <hip/hip_runtime.h>
#include <math.h>

// SimpleResBlock with AFNO2D for MI455X (gfx1250), fp32 end-to-end.
// Pipeline: conv3x3+bias(+BN1 stats) -> BN1+ReLU+row-rFFT -> col-FFT ->
//           WMMA block-diag complex MLP + softshrink + spectral gate ->
//           inv col-FFT -> inv row-FFT(+residual,+BN2 stats) -> ReLU(y+BN2(fo))

#define HH 512
#define WW 512
#define CC 32
#define NN 4
#define WF 257
#define LAM 0.01f
#define BN_EPS_ 1e-5f
#define RSQ512 0.04419417382415922f   // 1/sqrt(512)
#define CNT_NHW 1048576.0f            // N*H*W = 4*512*512
#define TOT (NN*CC*HH*WW)             // 33554432

typedef float v8f __attribute__((ext_vector_type(8)));
typedef float v2f __attribute__((ext_vector_type(2)));

__device__ __forceinline__ int rev9(int x) { return (int)(__brev((unsigned)x) >> 23); }

// 512-point in-place radix-2 DIF complex FFT in LDS, 256 threads.
// sign=+1: forward (e^{-i}); sign=-1: inverse twiddles (e^{+i}).
// Natural-order input; output bit-reversed (bin k lives at rev9(k)).
__device__ __forceinline__ void fft512_1(float* re, float* im, int tid, float sign) {
#pragma unroll
  for (int s = 0; s < 9; ++s) {
    int hs = 8 - s;
    int half = 1 << hs;
    int pos = tid & (half - 1);
    int i0 = ((tid >> hs) << (hs + 1)) + pos;
    int i1 = i0 + half;
    float ang = sign * -6.28318530717958647692f * (float)pos / (float)(half << 1);
    float sn, cs;
    __sincosf(ang, &sn, &cs);
    float ar = re[i0], ai = im[i0], br = re[i1], bi = im[i1];
    re[i0] = ar + br;
    im[i0] = ai + bi;
    float dr = ar - br, di = ai - bi;
    re[i1] = dr * cs - di * sn;
    im[i1] = dr * sn + di * cs;
    __syncthreads();
  }
}

__global__ void k_zero(float* stats) {
  if (threadIdx.x < 128) stats[threadIdx.x] = 0.f;
}

// ---- conv 3x3 pad 1 + bias; per-channel sum/sumsq atomics for BN1 ----
__global__ void k_conv(const float* __restrict__ x, const float* __restrict__ cw,
                       const float* __restrict__ cb, float* __restrict__ yraw,
                       float* __restrict__ stats) {
  __shared__ float wl[288];
  __shared__ float smS[8], smQ[8];
  int blk = blockIdx.x;
  int tile = blk & 1023;        // 1024 pixel-tiles per (b,oc) plane
  int oc = (blk >> 10) & 31;
  int b = blk >> 15;
  for (int t = threadIdx.x; t < 288; t += 256) wl[t] = cw[oc * 288 + t];
  __syncthreads();
  int p = (tile << 8) + threadIdx.x;
  int y0 = p >> 9, x0 = p & 511;
  const float* xb = x + (size_t)b * CC * HH * WW;
  __builtin_prefetch(xb + (size_t)(CC - 1) * HH * WW + (size_t)y0 * WW + x0, 0, 0);
  float acc = cb[oc];
#pragma unroll
  for (int ky = 0; ky < 3; ++ky) {
    int yy = y0 + ky - 1;
    if ((unsigned)yy < HH) {
#pragma unroll
      for (int kx = 0; kx < 3; ++kx) {
        int xx = x0 + kx - 1;
        if ((unsigned)xx < WW) {
          const float* xp = xb + (size_t)yy * WW + xx;
          const float* wp = wl + ky * 3 + kx;
#pragma unroll
          for (int ic = 0; ic < CC; ++ic)
            acc = fmaf(xp[(size_t)ic * HH * WW], wp[ic * 9], acc);
        }
      }
    }
  }
  yraw[(size_t)(b * CC + oc) * HH * WW + p] = acc;
  float s = acc, q = acc * acc;
  for (int o = 16; o > 0; o >>= 1) { s += __shfl_down(s, o, 32); q += __shfl_down(q, o, 32); }
  int lane = threadIdx.x & 31, wv = threadIdx.x >> 5;
  if (lane == 0) { smS[wv] = s; smQ[wv] = q; }
  __syncthreads();
  if (threadIdx.x == 0) {
    float ts = 0.f, tq = 0.f;
    for (int i = 0; i < 8; ++i) { ts += smS[i]; tq += smQ[i]; }
    atomicAdd(&stats[oc], ts);
    atomicAdd(&stats[32 + oc], tq);
  }
}

// ---- BN1 + ReLU fused with forward row rFFT; writes residual y and bins 0..256 ----
__global__ void k_rowfft_fwd(const float* __restrict__ yraw, const float* __restrict__ stats,
                             const float* __restrict__ g, const float* __restrict__ bt,
                             float* __restrict__ y, float2* __restrict__ X) {
  __shared__ float re[512], im[512];
  int row = blockIdx.x;                 // (b*C + c)*H + h
  int c = (row >> 9) & 31;
  int tid = threadIdx.x;
  float mean = stats[c] / CNT_NHW;
  float var = stats[32 + c] / CNT_NHW - mean * mean;
  float rs = rsqrtf(var + BN_EPS_);
  float gg = g[c], bb = bt[c];
  size_t base = (size_t)row * WW;
  for (int i = tid; i < 512; i += 256) {
    float v = yraw[base + i];
    v = fmaxf(gg * (v - mean) * rs + bb, 0.f);
    y[base + i] = v;
    re[i] = v;
    im[i] = 0.f;
  }
  __syncthreads();
  fft512_1(re, im, tid, 1.0f);
  size_t xb = (size_t)row * WF;
  for (int k = tid; k < WF; k += 256) {
    int pp = rev9(k);
    X[xb + k] = make_float2(re[pp] * RSQ512, im[pp] * RSQ512);
  }
}

// ---- column FFT over h (8 adjacent wf columns per workgroup); sign=+1 fwd, -1 inv ----
__global__ void k_colfft(float2* __restrict__ X, float sign) {
  __shared__ float re[8 * 512], im[8 * 512];
  int blk = blockIdx.x;
  int tile = blk % 33;
  int bc = blk / 33;                    // b*C + c
  int wf0 = tile * 8;
  int tid = threadIdx.x;
  size_t base = (size_t)bc * HH * WF;
  for (int col = 0; col < 8; ++col) {
    int wf = wf0 + col;
    for (int h = tid; h < 512; h += 256) {
      float2 v = (wf < WF) ? X[base + (size_t)h * WF + wf] : make_float2(0.f, 0.f);
      re[col * 512 + h] = v.x;
      im[col * 512 + h] = v.y;
    }
  }
  __syncthreads();
#pragma unroll
  for (int s = 0; s < 9; ++s) {
    int hs = 8 - s;
    int half = 1 << hs;
    float mInv = 1.0f / (float)(half << 1);
    for (int j = tid; j < 2048; j += 256) {
      int col = j >> 8, k = j & 255;
      int pos = k & (half - 1);
      int i0 = col * 512 + ((k >> hs) << (hs + 1)) + pos;
      int i1 = i0 + half;
      float ang = sign * -6.28318530717958647692f * (float)pos * mInv;
      float sn, cs;
      __sincosf(ang, &sn, &cs);
      float ar = re[i0], ai = im[i0], br = re[i1], bi = im[i1];
      re[i0] = ar + br;
      im[i0] = ai + bi;
      float dr = ar - br, di = ai - bi;
      re[i1] = dr * cs - di * sn;
      im[i1] = dr * sn + di * cs;
    }
    __syncthreads();
  }
  for (int col = 0; col < 8; ++col) {
    int wf = wf0 + col;
    if (wf >= WF) break;
    for (int h = tid; h < 512; h += 256) {
      int pp = rev9(h);
      X[base + (size_t)h * WF + wf] =
          make_float2(re[col * 512 + pp] * RSQ512, im[col * 512 + pp] * RSQ512);
    }
  }
}

// ---- WMMA block-diagonal complex 2-layer MLP + softshrink + spectral gate ----
// One workgroup (4 waves) per (b,h). Each wave owns 16-wide wf tiles.
// 32x32 block-diag weights => only the two diagonal 16x16 quadrants are non-zero:
// per tile/quadrant do M=16(wf) x N=16(ch) x K=16(ch) via 4x V_WMMA_F32_16X16X4_F32.
__global__ void k_mlp(float2* __restrict__ X, const float* __restrict__ w1g,
                      const float* __restrict__ b1g, const float* __restrict__ w2g,
                      const float* __restrict__ b2g) {
  __shared__ float wm[6][2][16][16];    // {W1r, W1i, -W1i, W2r, W2i, -W2i}[quadrant]
  __shared__ float bs[4][32];           // b1r, b1i, b2r, b2i
  __shared__ float stXr[4][32][17], stXi[4][32][17];   // per-wave gate staging
  __shared__ float stOr[4][32][17], stOi[4][32][17];   // per-wave o1 / z staging

  int tid = threadIdx.x;
  for (int t = tid; t < 6 * 2 * 16 * 16; t += 128) {
    int cidx = t & 15, r = (t >> 4) & 15, q = (t >> 8) & 1, mat = t >> 9;
    int kb = r >> 2;
    float v = 0.f;
    if ((cidx >> 2) == kb) {
      const float* src = (mat < 3) ? w1g : w2g;
      int m3 = (mat < 3) ? mat : mat - 3;
      int comp = (m3 == 0) ? 0 : 1;
      v = src[comp * 128 + (q * 4 + kb) * 16 + (r & 3) * 4 + (cidx & 3)];
      if (m3 == 2) v = -v;
    }
    wm[mat][q][r][cidx] = v;
  }
  if (tid < 32) {
    bs[0][tid] = b1g[tid];
    bs[1][tid] = b1g[32 + tid];
    bs[2][tid] = b2g[tid];
    bs[3][tid] = b2g[32 + tid];
  }
  __syncthreads();

  int lane = tid & 31;
  int wv = tid >> 5;
  int n = lane & 15;
  int off = (lane & 16) ? 2 : 0;        // lanes 16..31 carry K+2 (f32 A/B layout)
  int wbase = (lane & 16) ? 8 : 0;      // acc rows M = j + 8*(lane>=16)
  int b = blockIdx.x >> 9, h = blockIdx.x & 511;
  size_t chStride = (size_t)HH * WF;
  size_t base = ((size_t)b * CC) * chStride + (size_t)h * WF;

  for (int tile = wv; tile < 17; tile += 4) {
    int wf0 = tile * 16;
    int wfA = wf0 + n;
    bool inr = (wfA < WF);

    // ---- layer 1 (per quadrant): o1 = relu(Xr*W1r - Xi*W1i + b1r, Xi*W1r + Xr*W1i + b1i)
#pragma unroll
    for (int q = 0; q < 2; ++q) {
      v2f ar[4], ai[4];
#pragma unroll
      for (int kc = 0; kc < 4; ++kc) {
        int ch = q * 16 + kc * 4 + off;
        float2 f0 = inr ? X[base + (size_t)ch * chStride + wfA] : make_float2(0.f, 0.f);
        float2 f1 = inr ? X[base + (size_t)(ch + 1) * chStride + wfA] : make_float2(0.f, 0.f);
        ar[kc][0] = f0.x; ar[kc][1] = f1.x;
        ai[kc][0] = f0.y; ai[kc][1] = f1.y;
        stXr[wv][ch][n] = f0.x; stXr[wv][ch + 1][n] = f1.x;
        stXi[wv][ch][n] = f0.y; stXi[wv][ch + 1][n] = f1.y;
      }
      float b1r = bs[0][q * 16 + n], b1i = bs[1][q * 16 + n];
      v8f o1r = {b1r, b1r, b1r, b1r, b1r, b1r, b1r, b1r};
      v8f o1i = {b1i, b1i, b1i, b1i, b1i, b1i, b1i, b1i};
#pragma unroll
      for (int kc = 0; kc < 4; ++kc) {
        v2f m1r = {wm[0][q][kc * 4 + off][n], wm[0][q][kc * 4 + off + 1][n]};
        v2f m1i = {wm[1][q][kc * 4 + off][n], wm[1][q][kc * 4 + off + 1][n]};
        v2f m1n = {wm[2][q][kc * 4 + off][n], wm[2][q][kc * 4 + off + 1][n]};
        o1r = __builtin_amdgcn_wmma_f32_16x16x4_f32(false, ar[kc], false, m1r, (short)0, o1r, false, false);
        o1r = __builtin_amdgcn_wmma_f32_16x16x4_f32(false, ai[kc], false, m1n, (short)0, o1r, false, false);
        o1i = __builtin_amdgcn_wmma_f32_16x16x4_f32(false, ai[kc], false, m1r, (short)0, o1i, false, false);
        o1i = __builtin_amdgcn_wmma_f32_16x16x4_f32(false, ar[kc], false, m1i, (short)0, o1i, false, false);
      }
      int chS = q * 16 + n;
#pragma unroll
      for (int j = 0; j < 8; ++j) {
        stOr[wv][chS][wbase + j] = fmaxf(o1r[j], 0.f);
        stOi[wv][chS][wbase + j] = fmaxf(o1i[j], 0.f);
      }
    }
    asm volatile("s_wait_dscnt 0" ::: "memory");   // wave-local LDS RAW fence

    // ---- layer 2 + softshrink + gate (z = softshrink(o2) * X), staged back to stO
#pragma unroll
    for (int q = 0; q < 2; ++q) {
      v2f a2r[4], a2i[4];
#pragma unroll
      for (int kc = 0; kc < 4; ++kc) {
        int ch = q * 16 + kc * 4 + off;
        a2r[kc][0] = stOr[wv][ch][n]; a2r[kc][1] = stOr[wv][ch + 1][n];
        a2i[kc][0] = stOi[wv][ch][n]; a2i[kc][1] = stOi[wv][ch + 1][n];
      }
      float b2r = bs[2][q * 16 + n], b2i = bs[3][q * 16 + n];
      v8f o2r = {b2r, b2r, b2r, b2r, b2r, b2r, b2r, b2r};
      v8f o2i = {b2i, b2i, b2i, b2i, b2i, b2i, b2i, b2i};
#pragma unroll
      for (int kc = 0; kc < 4; ++kc) {
        v2f m2r = {wm[3][q][kc * 4 + off][n], wm[3][q][kc * 4 + off + 1][n]};
        v2f m2i = {wm[4][q][kc * 4 + off][n], wm[4][q][kc * 4 + off + 1][n]};
        v2f m2n = {wm[5][q][kc * 4 + off][n], wm[5][q][kc * 4 + off + 1][n]};
        o2r = __builtin_amdgcn_wmma_f32_16x16x4_f32(false, a2r[kc], false, m2r, (short)0, o2r, false, false);
        o2r = __builtin_amdgcn_wmma_f32_16x16x4_f32(false, a2i[kc], false, m2n, (short)0, o2r, false, false);
        o2i = __builtin_amdgcn_wmma_f32_16x16x4_f32(false, a2i[kc], false, m2r, (short)0, o2i, false, false);
        o2i = __builtin_amdgcn_wmma_f32_16x16x4_f32(false, a2r[kc], false, m2i, (short)0, o2i, false, false);
      }
      int chS = q * 16 + n;
#pragma unroll
      for (int j = 0; j < 8; ++j) {
        float vr = o2r[j];
        vr = (vr > LAM) ? vr - LAM : ((vr < -LAM) ? vr + LAM : 0.f);
        float vi = o2i[j];
        vi = (vi > LAM) ? vi - LAM : ((vi < -LAM) ? vi + LAM : 0.f);
        float xr = stXr[wv][chS][wbase + j];
        float xi = stXi[wv][chS][wbase + j];
        stOr[wv][chS][wbase + j] = vr * xr - vi * xi;
        stOi[wv][chS][wbase + j] = vr * xi + vi * xr;
      }
    }
    asm volatile("s_wait_dscnt 0" ::: "memory");

    // ---- coalesced store of gated spectrum
    if (inr) {
#pragma unroll
      for (int cp = 0; cp < 16; ++cp) {
        int ch = cp * 2 + ((lane & 16) ? 1 : 0);
        X[base + (size_t)ch * chStride + wfA] = make_float2(stOr[wv][ch][n], stOi[wv][ch][n]);
      }
    }
  }
}

// ---- inverse row FFT (Hermitian c2r) + residual add; BN2 stats atomics ----
__global__ void k_rowifft(const float2* __restrict__ X, const float* __restrict__ y,
                          float* __restrict__ fo, float* __restrict__ stats) {
  __shared__ float re[512], im[512];
  __shared__ float smS[8], smQ[8];
  int row = blockIdx.x;
  int c = (row >> 9) & 31;
  int tid = threadIdx.x;
  size_t xb = (size_t)row * WF;
  if (tid == 0) {
    float2 f0 = X[xb];       re[0] = f0.x;   im[0] = f0.y;
    float2 fN = X[xb + 256]; re[256] = fN.x; im[256] = fN.y;
  } else {
    float2 f = X[xb + tid];
    re[tid] = f.x;        im[tid] = f.y;
    re[512 - tid] = f.x;  im[512 - tid] = -f.y;
  }
  __syncthreads();
  fft512_1(re, im, tid, -1.0f);
  size_t base = (size_t)row * WW;
  float s = 0.f, q = 0.f;
  for (int i = tid; i < 512; i += 256) {
    float v = re[rev9(i)] * RSQ512 + y[base + i];
    fo[base + i] = v;
    s += v;
    q += v * v;
  }
  for (int o = 16; o > 0; o >>= 1) { s += __shfl_down(s, o, 32); q += __shfl_down(q, o, 32); }
  int lane = tid & 31, wv = tid >> 5;
  if (lane == 0) { smS[wv] = s; smQ[wv] = q; }
  __syncthreads();
  if (tid == 0) {
    float ts = 0.f, tq = 0.f;
    for (int i = 0; i < 8; ++i) { ts += smS[i]; tq += smQ[i]; }
    atomicAdd(&stats[64 + c], ts);
    atomicAdd(&stats[96 + c], tq);
  }
}

// ---- out = relu(y + BN2(fo)) ----
__global__ void k_final(const float* __restrict__ y, const float* __restrict__ fo,
                        const float* __restrict__ stats, const float* __restrict__ g2,
                        const float* __restrict__ bt2, float* __restrict__ out) {
  size_t i = (size_t)blockIdx.x * 256 + threadIdx.x;
  int c = (int)((i >> 18) & 31);
  float mean = stats[64 + c] / CNT_NHW;
  float var = stats[96 + c] / CNT_NHW - mean * mean;
  float rs = rsqrtf(var + BN_EPS_);
  float v = g2[c] * (fo[i] - mean) * rs + bt2[c];
  out[i] = fmaxf(y[i] + v, 0.f);
}

extern "C" void kernel_launch(void* const* d_in, const int* in_sizes, int n_in,
                              void* d_out, int out_size, void* d_ws, size_t ws_size,
                              hipStream_t stream) {
  (void)in_sizes; (void)n_in; (void)out_size; (void)ws_size;
  const float* x   = (const float*)d_in[0];
  const float* cw  = (const float*)d_in[1];
  const float* cb  = (const float*)d_in[2];
  const float* g1  = (const float*)d_in[3];
  const float* bt1 = (const float*)d_in[4];
  const float* w1  = (const float*)d_in[5];
  const float* b1  = (const float*)d_in[6];
  const float* w2  = (const float*)d_in[7];
  const float* b2  = (const float*)d_in[8];
  const float* g2  = (const float*)d_in[9];
  const float* bt2 = (const float*)d_in[10];
  float* out = (float*)d_out;

  char* ws = (char*)d_ws;
  float*  yraw  = (float*)(ws);                                   // 128 MiB; reused as fo
  float*  ybuf  = (float*)(ws + 134217728ull);                    // 128 MiB residual
  float2* X     = (float2*)(ws + 268435456ull);                   // 134.7 MB half-spectrum
  float*  stats = (float*)(ws + 268435456ull + 134742016ull);     // 128 floats

  k_zero<<<1, 128, 0, stream>>>(stats);
  k_conv<<<NN * CC * 1024, 256, 0, stream>>>(x, cw, cb, yraw, stats);
  k_rowfft_fwd<<<NN * CC * HH, 256, 0, stream>>>(yraw, stats, g1, bt1, ybuf, X);
  k_colfft<<<NN * CC * 33, 256, 0, stream>>>(X, 1.0f);
  k_mlp<<<NN * HH, 128, 0, stream>>>(X, w1, b1, w2, b2);
  k_colfft<<<NN * CC * 33, 256, 0, stream>>>(X, -1.0f);
  k_rowifft<<<NN * CC * HH, 256, 0, stream>>>(X, ybuf, yraw, stats);
  k_final<<<TOT / 256, 256, 0, stream>>>(ybuf, yraw, stats, g2, bt2, out);
}